// Prototype_74663711473993
// MI455X (gfx1250) — compile-verified
//
#include <hip/hip_runtime.h>
#include <hip/hip_bf16.h>

typedef float v2f __attribute__((ext_vector_type(2)));
typedef float v8f __attribute__((ext_vector_type(8)));

constexpr int NB = 32;            // batch
constexpr int NH = 56, NW = 56;   // spatial
constexpr int NC = 256;           // channels (GEMM K)
constexpr int NP = 128;           // prototypes (GEMM N)
constexpr int NS = NH * NW;       // 3136 spatial rows per batch
constexpr int MTILE = 64;         // spatial rows per batch-tile (4 strips of 16)
constexpr int NTILES = NP / 16;   // 8 column tiles
constexpr int TILES = NS / MTILE; // 49 (exact)
constexpr float MAXDIST = 100.0f * 256.0f; // 25600
constexpr float EPS = 1e-4f;

// ---------------------------------------------------------------------------
// Kernel 0: yTy[p] = sum_c proto[c][p]^2 ; init (B,P) min/max atomic buffers.
// ---------------------------------------------------------------------------
__global__ __launch_bounds__(128)
void proto_init(const float* __restrict__ proto,
                unsigned int* __restrict__ wmin,
                unsigned int* __restrict__ wmax,
                float* __restrict__ yty)
{
    int g = blockIdx.x * 128 + threadIdx.x;
    if (g < NB * NP) {
        wmin[g] = 0x7F800000u;  // +inf bits (values are non-negative floats)
        wmax[g] = 0u;           // 0.0f bits
    }
    if (blockIdx.x == 0) {
        int p = threadIdx.x;
        float s = 0.f;
        for (int c = 0; c < NC; ++c) {
            float v = proto[c * NP + p];
            s += v * v;
        }
        yty[p] = s;
    }
}

// ---------------------------------------------------------------------------
// Kernel 1: fused GEMM (WMMA f32 16x16x4) + distance/similarity epilogue +
// spatial min/max reduction into (B,P) workspace via u32 atomics.
//   Block: 256 threads = 8 waves. blockIdx.x = spatial tile, blockIdx.y = batch pair.
//   Waves 0-3 -> batch b0, waves 4-7 -> batch b0+1; each wave owns a 16-row strip.
//   LDS: A tiles (2 x 64 x 256 f32) + prototypes in interleaved fragment layout
//        (element (k,p) at ((k>>1)*NP+p)*2 + (k&1)) + xTx[128].
// ---------------------------------------------------------------------------
__global__ __launch_bounds__(256)
void proto_main(const float* __restrict__ in,
                const float* __restrict__ proto,
                const float* __restrict__ lscale,
                const float* __restrict__ yty,
                unsigned int* __restrict__ wmin,
                unsigned int* __restrict__ wmax)
{
    extern __shared__ float lds[];
    float* a_lds = lds;                        // 2*MTILE*NC = 32768 floats
    float* p_lds = lds + 2 * MTILE * NC;       // NC*NP      = 32768 floats (interleaved)
    float* xtx   = p_lds + NC * NP;            // 2*MTILE    = 128 floats

    const int tile = blockIdx.x;               // 0..48
    const int b0   = blockIdx.y * 2;           // batches b0, b0+1
    const int tid  = threadIdx.x;

    // ---- stage A tiles (contiguous 64 KB per batch) ----
    for (int bb = 0; bb < 2; ++bb) {
        const float4* src = (const float4*)(in + ((size_t)(b0 + bb) * NS + (size_t)tile * MTILE) * NC);
        float4* dst = (float4*)(a_lds + bb * MTILE * NC);
        for (int i = tid; i < MTILE * NC / 4; i += 256) dst[i] = src[i];
    }
    // ---- stage prototypes into interleaved fragment layout ----
    // v2f element index ((k>>1)*NP + p) holds { proto[2*kp][p], proto[2*kp+1][p] }
    {
        float2* dst = (float2*)p_lds;
        for (int i = tid; i < (NC / 2) * NP; i += 256) {
            int kp = i >> 7;          // i / NP
            int p  = i & (NP - 1);
            float v0 = proto[(2 * kp) * NP + p];
            float v1 = proto[(2 * kp + 1) * NP + p];
            dst[i] = make_float2(v0, v1);
        }
    }
    __syncthreads();

    // ---- xTx per local row (128 rows across both batches) ----
    if (tid < 2 * MTILE) {
        const float4* row = (const float4*)(a_lds + tid * NC);
        float s = 0.f;
        for (int i = 0; i < NC / 4; ++i) {
            float4 v = row[i];
            s += v.x * v.x + v.y * v.y + v.z * v.z + v.w * v.w;
        }
        xtx[tid] = s;
    }
    __syncthreads();

    const int wave  = tid >> 5;        // 0..7
    const int lane  = tid & 31;
    const int lm    = lane & 15;
    const int hsel  = lane >> 4;       // 0: lanes 0-15, 1: lanes 16-31
    const int bb    = wave >> 2;       // batch select within block
    const int strip = wave & 3;        // 16-row strip within MTILE

    const float* Arow = a_lds + (bb * MTILE + strip * 16) * NC;

    v8f acc[NTILES];
    #pragma unroll
    for (int n = 0; n < NTILES; ++n) acc[n] = (v8f){0.f, 0.f, 0.f, 0.f, 0.f, 0.f, 0.f, 0.f};

    // A fragment (16x4 f32): lanes 0-15 hold K={k0,k0+1} of row M=lane,
    // lanes 16-31 hold K={k0+2,k0+3} of row M=lane-16.  Contiguous -> one b64.
    const int a_off = lm * NC + 2 * hsel;
    const v2f* Pfrag = (const v2f*)p_lds;

    #pragma unroll 1
    for (int k0 = 0; k0 < NC; k0 += 4) {
        v2f Af = *(const v2f*)(Arow + a_off + k0);
        // B fragment: rows kA=k0+2*hsel and kA+1, col p=n*16+lm -> one b64 in
        // the interleaved layout at v2f index ((k0>>1)+hsel)*NP + p.
        const v2f* Bp = Pfrag + ((k0 >> 1) + hsel) * NP + lm;
        #pragma unroll
        for (int n = 0; n < NTILES; ++n) {
            v2f Bf = Bp[n * 16];
            acc[n] = __builtin_amdgcn_wmma_f32_16x16x4_f32(
                false, Af, false, Bf, (short)0, acc[n], false, false);
        }
    }

    // ---- epilogue: D tile element (vgpr r, lane) = (row r + 8*hsel, col n*16+lm)
    // Fully unrolled so acc[][] indexing is static (no scratch demotion).
    const int row_loc = bb * MTILE + strip * 16 + hsel * 8;   // + r -> xtx index
    const int s_base  = tile * MTILE + strip * 16 + hsel * 8; // + r -> spatial index
    const int bglob   = b0 + bb;

    #pragma unroll
    for (int n = 0; n < NTILES; ++n) {
        const int p  = n * 16 + lm;
        const float yv = yty[p];
        float mn = 3.0e38f;
        float mx = 0.f;
        #pragma unroll
        for (int r = 0; r < 8; ++r) {
            float xt  = xtx[row_loc + r];
            float ls  = lscale[(size_t)(s_base + r) * NP + p];
            float sf  = __expf(ls);
            float xty = acc[n][r];
            float nsq = fmaxf(xt - 2.f * xty + yv, 0.f);
            float dmin = nsq / (sf + EPS);
            float sim  = __logf((nsq + 1.f) / (nsq + EPS)) * sf;
            mn = fminf(mn, dmin);
            mx = fmaxf(mx, sim);
        }
        // combine lane halves (M=r vs M=r+8 live in opposite 16-lane halves)
        mn = fminf(mn, __shfl_xor(mn, 16));
        mx = fmaxf(mx, __shfl_xor(mx, 16));
        if (lane < 16) {
            atomicMin(wmin + bglob * NP + p, __float_as_uint(mn));
            atomicMax(wmax + bglob * NP + p, __float_as_uint(mx));
        }
    }
}

// ---------------------------------------------------------------------------
// Kernel 2: emit max_similarity_scores (B,P) + cluster/separation costs.
// ---------------------------------------------------------------------------
__global__ __launch_bounds__(128)
void proto_final(const unsigned int* __restrict__ wmin,
                 const unsigned int* __restrict__ wmax,
                 const float* __restrict__ pcc,
                 float* __restrict__ out)
{
    __shared__ float r1[128];
    __shared__ float r2[128];
    const int t = threadIdx.x;

    for (int i = t; i < NB * NP; i += 128)
        out[i] = __uint_as_float(wmax[i]);

    float s1 = 0.f, s2 = 0.f;
    for (int b = 0; b < NB; ++b) {
        float md = __uint_as_float(wmin[b * NP + t]);
        float v  = MAXDIST - md;
        float m  = pcc[b * NP + t];
        r1[t] = v * m;
        r2[t] = v * (1.f - m);
        __syncthreads();
        for (int st = 64; st > 0; st >>= 1) {
            if (t < st) {
                r1[t] = fmaxf(r1[t], r1[t + st]);
                r2[t] = fmaxf(r2[t], r2[t + st]);
            }
            __syncthreads();
        }
        if (t == 0) { s1 += MAXDIST - r1[0]; s2 += MAXDIST - r2[0]; }
        __syncthreads();
    }
    if (t == 0) {
        out[NB * NP]     = s1 / (float)NB;
        out[NB * NP + 1] = s2 / (float)NB;
    }
}

// ---------------------------------------------------------------------------
extern "C" void kernel_launch(void* const* d_in, const int* in_sizes, int n_in,
                              void* d_out, int out_size, void* d_ws, size_t ws_size,
                              hipStream_t stream)
{
    const float* inputs = (const float*)d_in[0];   // (32,56,56,256)
    const float* pcc    = (const float*)d_in[1];   // (32,128)
    const float* proto  = (const float*)d_in[2];   // (1,1,256,128)
    const float* lscale = (const float*)d_in[3];   // (56,56,128)
    float* out = (float*)d_out;                    // 4096 + 2

    unsigned int* wmin = (unsigned int*)d_ws;      // (B,P) min bits
    unsigned int* wmax = wmin + NB * NP;           // (B,P) max bits
    float* yty = (float*)(wmax + NB * NP);         // (P,)

    proto_init<<<dim3((NB * NP) / 128), dim3(128), 0, stream>>>(proto, wmin, wmax, yty);

    const size_t shmem = (size_t)(2 * MTILE * NC + NC * NP + 2 * MTILE) * sizeof(float);
    proto_main<<<dim3(TILES, NB / 2), dim3(256), shmem, stream>>>(
        inputs, proto, lscale, yty, wmin, wmax);

    proto_final<<<dim3(1), dim3(128), 0, stream>>>(wmin, wmax, pcc, out);
}